// DatasetIndexedTopK_24429773980154
// MI455X (gfx1250) — compile-verified
//
#include <hip/hip_runtime.h>

typedef float v2f __attribute__((ext_vector_type(2)));
typedef float v8f __attribute__((ext_vector_type(8)));

#define QT       16     // queries per block
#define CHUNK    4096   // candidates per block
#define QUARTER  1024   // candidates per GEMM/selection round
#define THREADS  256    // 8 waves (wave32)
#define KMAX     128    // max supported k
#define LISTCAP  256    // running list capacity (>= 2*KMAX)
#define QSTRIDE  130    // padded LDS row stride (floats): 8B aligned, no bank conflicts

// Monotone key transform: float total order -> unsigned ascending order.
__device__ __forceinline__ unsigned fkey(float f) {
  unsigned u = __float_as_uint(f);
  return (u & 0x80000000u) ? ~u : (u | 0x80000000u);
}

// Parallel leader step of one radix pass: 16 threads/query reduce 16-bin
// groups, then the per-query leader scans 16 group sums + 16 bins (32 iters
// instead of 256). Folds chosen bin into prefix, updates remaining-k.
__device__ __forceinline__ void leader_scan(int tid, const int* partic,
                                            const unsigned* hist, int* gsum,
                                            unsigned* sel_prefix, int* sel_remk) {
  const int qg = tid & 15, g = tid >> 4;
  int gs = 0;
  #pragma unroll
  for (int b = 0; b < 16; ++b) gs += (int)hist[qg * 256 + g * 16 + b];
  gsum[qg * 16 + g] = gs;
  __syncthreads();
  if (tid < QT && partic[tid]) {
    int remk = sel_remk[tid];
    int G = 0;
    for (int g2 = 15; g2 >= 0; --g2) {
      int c = gsum[tid * 16 + g2];
      if (c >= remk) { G = g2; break; }
      remk -= c;
    }
    unsigned bin = (unsigned)(G * 16);
    for (int b = G * 16 + 15; b >= G * 16; --b) {
      int c = (int)hist[tid * 256 + b];
      if (c >= remk) { bin = (unsigned)b; break; }
      remk -= c;
    }
    sel_prefix[tid] = (sel_prefix[tid] << 8) | bin;
    sel_remk[tid] = remk;
  }
  __syncthreads();
}

// 4-pass radix select over each thread's 64 in-register WMMA scores.
__device__ __forceinline__ void reg_select_passes(const v8f* sc, int tid, int qloc,
                                                  const int* partic, unsigned* hist,
                                                  int* gsum, unsigned* sel_prefix,
                                                  int* sel_remk) {
  for (int p = 0; p < 4; ++p) {
    for (int i = tid; i < QT * 256; i += THREADS) hist[i] = 0u;
    __syncthreads();
    if (partic[qloc]) {
      unsigned pref = sel_prefix[qloc];
      int shift = 24 - 8 * p;
      #pragma unroll
      for (int t = 0; t < 8; ++t)
        #pragma unroll
        for (int e = 0; e < 8; ++e) {
          unsigned key = fkey(sc[t][e]);
          bool in = (p == 0) || ((key >> (shift + 8)) == pref);
          if (in) atomicAdd(&hist[qloc * 256 + ((key >> shift) & 255u)], 1u);
        }
    }
    __syncthreads();
    leader_scan(tid, partic, hist, gsum, sel_prefix, sel_remk);
  }
}

// Gather exactly k entries (greater + ticketed equals) from registers to list.
__device__ __forceinline__ void reg_gather(const v8f* sc, int wave, int h, int qloc,
                                           const int* partic, const unsigned* sel_prefix,
                                           const int* sel_remk, int* eq_n, int* list_n,
                                           float* list_s, int* list_i,
                                           const int* ibase, int qn) {
  if (!partic[qloc]) return;
  unsigned kth = sel_prefix[qloc];
  int r = sel_remk[qloc];
  #pragma unroll
  for (int t = 0; t < 8; ++t)
    #pragma unroll
    for (int e = 0; e < 8; ++e) {
      float f = sc[t][e];
      unsigned key = fkey(f);
      int take = 0;
      if (key > kth) take = 1;
      else if (key == kth) { int tk = atomicAdd(&eq_n[qloc], 1); if (tk < r) take = 1; }
      if (take) {
        int slot = atomicAdd(&list_n[qloc], 1);
        int pos  = qn * QUARTER + wave * 128 + t * 16 + h * 8 + e;
        list_s[qloc * LISTCAP + slot] = f;
        list_i[qloc * LISTCAP + slot] = ibase[pos];
      }
    }
}

// 4-pass radix select over the LDS running list (16 threads per query).
__device__ __forceinline__ void list_select_passes(int tid, const int* partic,
                                                   const float* list_s, const int* list_n,
                                                   unsigned* hist, int* gsum,
                                                   unsigned* sel_prefix, int* sel_remk) {
  const int qg = tid & 15, g = tid >> 4;
  const int m = list_n[qg];
  for (int p = 0; p < 4; ++p) {
    for (int i = tid; i < QT * 256; i += THREADS) hist[i] = 0u;
    __syncthreads();
    if (partic[qg]) {
      unsigned pref = sel_prefix[qg];
      int shift = 24 - 8 * p;
      for (int j = g; j < m; j += 16) {
        unsigned key = fkey(list_s[qg * LISTCAP + j]);
        bool in = (p == 0) || ((key >> (shift + 8)) == pref);
        if (in) atomicAdd(&hist[qg * 256 + ((key >> shift) & 255u)], 1u);
      }
    }
    __syncthreads();
    leader_scan(tid, partic, hist, gsum, sel_prefix, sel_remk);
  }
}

// Compact participating queries' lists back to exactly k; refresh threshold.
__device__ __forceinline__ void list_compact(int tid, int k, const int* partic,
                                             float* list_s, int* list_i, int* list_n,
                                             const unsigned* sel_prefix, const int* sel_remk,
                                             int* eq_n, int* out_n,
                                             float* tmp_s, int* tmp_i, unsigned* thr_key) {
  const int qg = tid & 15, g = tid >> 4;
  const int m = list_n[qg];
  if (partic[qg]) {
    unsigned kth = sel_prefix[qg];
    int r = sel_remk[qg];
    for (int j = g; j < m; j += 16) {
      float f = list_s[qg * LISTCAP + j];
      unsigned key = fkey(f);
      int take = 0;
      if (key > kth) take = 1;
      else if (key == kth) { int tk = atomicAdd(&eq_n[qg], 1); if (tk < r) take = 1; }
      if (take) {
        int slot = atomicAdd(&out_n[qg], 1);
        tmp_s[qg * KMAX + slot] = f;
        tmp_i[qg * KMAX + slot] = list_i[qg * LISTCAP + j];
      }
    }
  }
  __syncthreads();
  if (partic[qg]) {
    for (int j = g; j < k; j += 16) {
      list_s[qg * LISTCAP + j] = tmp_s[qg * KMAX + j];
      list_i[qg * LISTCAP + j] = tmp_i[qg * KMAX + j];
    }
  }
  if (tid < QT && partic[tid]) { list_n[tid] = k; thr_key[tid] = sel_prefix[tid]; }
  __syncthreads();
}

// Kernel 1: fp32 WMMA scores for a 16-query x 4096-candidate panel + exact
// per-query top-k over the panel, written to workspace partial lists.
// Quarter 0 does a full register radix-select; quarters 1..3 use the known
// k-th threshold as a pure-VALU filter and only compact when survivors exist.
__global__ __launch_bounds__(THREADS)
void gemm_chunk_topk(const float* __restrict__ qemb,
                     const float* __restrict__ cemb,
                     const int*   __restrict__ cidx,
                     float* __restrict__ ws_s,
                     int*   __restrict__ ws_i,
                     int nch, int k)
{
  __shared__ float    qbuf[QT * QSTRIDE];
  __shared__ unsigned hist[QT * 256];
  __shared__ int      gsum[QT * 16];
  __shared__ float    list_s[QT * LISTCAP];
  __shared__ int      list_i[QT * LISTCAP];
  __shared__ float    tmp_s[QT * KMAX];
  __shared__ int      tmp_i[QT * KMAX];
  __shared__ int      list_n[QT];
  __shared__ unsigned sel_prefix[QT];
  __shared__ int      sel_remk[QT];
  __shared__ int      eq_n[QT];
  __shared__ int      out_n[QT];
  __shared__ int      scnt[QT];
  __shared__ int      needsel[QT];
  __shared__ int      partic[QT];
  __shared__ unsigned thr_key[QT];
  __shared__ int      any_flag;

  const int tid   = threadIdx.x;
  const int wave  = tid >> 5;
  const int lane  = tid & 31;
  const int h     = lane >> 4;   // half-wave: K-pair / M-half per WMMA layout
  const int qloc  = lane & 15;   // this lane's query column (D-matrix N = lane%16)
  const int qtile = blockIdx.x;
  const int chunk = blockIdx.y;

  for (int i = tid; i < QT * 128; i += THREADS) {
    int r = i >> 7, c = i & 127;
    qbuf[r * QSTRIDE + c] = qemb[(size_t)(qtile * QT + r) * 128 + c];
  }
  if (tid < QT) list_n[tid] = 0;
  __syncthreads();

  const float* cbase = cemb + (size_t)chunk * CHUNK * 128;
  const int*   ibase = cidx + (size_t)chunk * CHUNK;

  for (int qn = 0; qn < 4; ++qn) {
    // ---- GEMM quarter: this wave computes 128 candidates x 16 queries.
    // A (16x4 f32): lane m=lane&15 -> candidate row, VGPR pair = cols kk+2h..+1.
    // B (4x16 f32): lane n=lane&15 -> query col,    VGPR pair = rows kk+2h..+1.
    // D (16x16 f32): lane owns query qloc, 8 candidate rows M = 8h+e.
    v8f sc[8];
    #pragma unroll
    for (int t = 0; t < 8; ++t) {
      int crow = qn * QUARTER + wave * 128 + t * 16 + qloc;
      const float* ap = cbase + (size_t)crow * 128 + 2 * h;
      const float* bp = &qbuf[qloc * QSTRIDE + 2 * h];
      v8f acc = {};
      #pragma unroll
      for (int kk = 0; kk < 32; ++kk) {
        v2f a = *(const v2f*)(ap + 4 * kk);
        v2f b = *(const v2f*)(bp + 4 * kk);
        acc = __builtin_amdgcn_wmma_f32_16x16x4_f32(false, a, false, b,
                                                    (short)0, acc, false, false);
      }
      sc[t] = acc;
    }
    __syncthreads();

    if (qn == 0) {
      // Full exact top-k of the first 1024 per query.
      if (tid < QT) { partic[tid] = 1; sel_prefix[tid] = 0u; sel_remk[tid] = k; eq_n[tid] = 0; }
      __syncthreads();
      reg_select_passes(sc, tid, qloc, partic, hist, gsum, sel_prefix, sel_remk);
      reg_gather(sc, wave, h, qloc, partic, sel_prefix, sel_remk,
                 eq_n, list_n, list_s, list_i, ibase, qn);
      __syncthreads();
      if (tid < QT) thr_key[tid] = sel_prefix[tid];
      __syncthreads();
    } else {
      // Threshold filter (no atomics): count survivors per query.
      if (tid < QT) scnt[tid] = 0;
      __syncthreads();
      unsigned thr = thr_key[qloc];
      int loc = 0;
      #pragma unroll
      for (int t = 0; t < 8; ++t)
        #pragma unroll
        for (int e = 0; e < 8; ++e)
          if (fkey(sc[t][e]) > thr) ++loc;
      if (loc) atomicAdd((unsigned*)&scnt[qloc], 1u * loc);
      __syncthreads();
      if (tid < QT) needsel[tid] = (list_n[tid] + scnt[tid] > LISTCAP) ? 1 : 0;
      __syncthreads();
      if (tid == 0) {
        int a = 0;
        for (int q = 0; q < QT; ++q) a |= needsel[q];
        any_flag = a;
      }
      // Direct path: append survivors of non-overflowing queries.
      if (!needsel[qloc]) {
        #pragma unroll
        for (int t = 0; t < 8; ++t)
          #pragma unroll
          for (int e = 0; e < 8; ++e) {
            float f = sc[t][e];
            if (fkey(f) > thr) {
              int slot = atomicAdd(&list_n[qloc], 1);
              int pos  = qn * QUARTER + wave * 128 + t * 16 + h * 8 + e;
              list_s[qloc * LISTCAP + slot] = f;
              list_i[qloc * LISTCAP + slot] = ibase[pos];
            }
          }
      }
      __syncthreads();
      // Rare fallback: full masked select for queries whose survivors overflow.
      if (any_flag) {
        if (tid < QT && needsel[tid]) { sel_prefix[tid] = 0u; sel_remk[tid] = k; eq_n[tid] = 0; }
        __syncthreads();
        reg_select_passes(sc, tid, qloc, needsel, hist, gsum, sel_prefix, sel_remk);
        reg_gather(sc, wave, h, qloc, needsel, sel_prefix, sel_remk,
                   eq_n, list_n, list_s, list_i, ibase, qn);
        __syncthreads();
      }
      // Compact any query whose list grew beyond k; refresh its threshold.
      if (tid < QT) {
        partic[tid] = (list_n[tid] > k) ? 1 : 0;
        sel_prefix[tid] = 0u; sel_remk[tid] = k; eq_n[tid] = 0; out_n[tid] = 0;
      }
      __syncthreads();
      if (tid == 0) {
        int a = 0;
        for (int q = 0; q < QT; ++q) a |= partic[q];
        any_flag = a;
      }
      __syncthreads();
      if (any_flag) {
        list_select_passes(tid, partic, list_s, list_n, hist, gsum, sel_prefix, sel_remk);
        list_compact(tid, k, partic, list_s, list_i, list_n,
                     sel_prefix, sel_remk, eq_n, out_n, tmp_s, tmp_i, thr_key);
      }
    }
  }

  // Emit: each query's list holds exactly k entries (unsorted).
  {
    const int qg = tid & 15, g = tid >> 4;
    size_t obase = ((size_t)(qtile * QT + qg) * (size_t)nch + (size_t)chunk) * (size_t)k;
    for (int j = g; j < k; j += 16) {
      ws_s[obase + j] = list_s[qg * LISTCAP + j];
      ws_i[obase + j] = list_i[qg * LISTCAP + j];
    }
  }
}

// Kernel 2: per-query merge of nch partial lists -> sorted-descending top-k.
// Cached per-thread argmax; only the winning thread's stripe is rescanned.
__global__ __launch_bounds__(256)
void merge_topk(float* __restrict__ ws_s, const int* __restrict__ ws_i,
                float* __restrict__ out_s, int* __restrict__ out_i,
                int nch, int k)
{
  __shared__ float bv[256];
  __shared__ int   bp[256];
  const int tid = threadIdx.x;
  const int q   = blockIdx.x;
  const int m   = nch * k;
  float*      s   = ws_s + (size_t)q * m;
  const int*  ids = ws_i + (size_t)q * m;

  float mv = -INFINITY; int mp = 0x7fffffff;
  for (int j = tid; j < m; j += 256) {
    float v = s[j];
    if (v > mv) { mv = v; mp = j; }
  }
  for (int it = 0; it < k; ++it) {
    bv[tid] = mv; bp[tid] = mp;
    __syncthreads();
    for (int off = 128; off > 0; off >>= 1) {
      if (tid < off) {
        float v2 = bv[tid + off]; int p2 = bp[tid + off];
        if (v2 > bv[tid] || (v2 == bv[tid] && p2 < bp[tid])) { bv[tid] = v2; bp[tid] = p2; }
      }
      __syncthreads();
    }
    float wv = bv[0]; int wp = bp[0];
    if (tid == 0) {
      out_s[(size_t)q * k + it] = wv;
      out_i[(size_t)q * k + it] = ids[wp];
    }
    __syncthreads();
    if ((wp & 255) == tid) {        // owner: retire winner, rescan own stripe
      s[wp] = -INFINITY;
      mv = -INFINITY; mp = 0x7fffffff;
      for (int j = tid; j < m; j += 256) {
        float v = s[j];
        if (v > mv) { mv = v; mp = j; }
      }
    }
    __syncthreads();
  }
}

extern "C" void kernel_launch(void* const* d_in, const int* in_sizes, int n_in,
                              void* d_out, int out_size, void* d_ws, size_t ws_size,
                              hipStream_t stream) {
  const float* qemb = (const float*)d_in[0];   // [Q,128] f32
  const float* cemb = (const float*)d_in[1];   // [N,128] f32 (flattened NB*CB)
  const int*   cidx = (const int*)d_in[2];     // [N] int32
  (void)n_in; (void)ws_size;

  const int Q   = in_sizes[0] / 128;
  const int N   = in_sizes[2];
  const int nch = N / CHUNK;
  const int k   = out_size / (2 * Q);          // tuple (scores, ids) concat flat

  float* ws_s  = (float*)d_ws;                              // [Q][nch][k] f32
  int*   ws_i  = (int*)(ws_s + (size_t)Q * nch * k);        // [Q][nch][k] i32
  float* out_s = (float*)d_out;                             // [Q][k] f32
  int*   out_i = (int*)(out_s + (size_t)Q * k);             // [Q][k] i32 (raw bits)

  dim3 g1(Q / QT, nch);   // q-tile fastest -> chunk stays hot in L2 across q-tiles
  hipLaunchKernelGGL(gemm_chunk_topk, g1, dim3(THREADS), 0, stream,
                     qemb, cemb, cidx, ws_s, ws_i, nch, k);
  hipLaunchKernelGGL(merge_topk, dim3(Q), dim3(256), 0, stream,
                     ws_s, ws_i, out_s, out_i, nch, k);
}